// FastAttention_9500467659067
// MI455X (gfx1250) — compile-verified
//
#include <hip/hip_runtime.h>
#include <cstdint>

// ---------------- WMMA f32 16x16x4 helpers (wave32, gfx1250) ----------------
typedef __attribute__((ext_vector_type(2))) float v2f;
typedef __attribute__((ext_vector_type(8))) float v8f;
typedef __attribute__((ext_vector_type(4))) int v4i;

__device__ __forceinline__ v8f wmma4(v2f a, v2f b, v8f c) {
  // D = A(16x4) * B(4x16) + C(16x16), exact f32
  return __builtin_amdgcn_wmma_f32_16x16x4_f32(false, a, false, b, (short)0, c,
                                               false, false);
}

// A fragment: A[m][k], row-major source, lda elements per row.
// lanes 0-15: K = k0,k0+1 ; lanes 16-31: K = k0+2,k0+3 ; M = lane&15
__device__ __forceinline__ v2f frag_a(const float* A, int lda, int m0, int k0) {
  int lane = threadIdx.x & 31;
  int half = lane >> 4, m = lane & 15;
  const float* p = A + (m0 + m) * lda + k0 + half * 2;
  v2f r; r.x = p[0]; r.y = p[1]; return r;
}

// A fragment from transposed source: A[m][k] = M[k][m] (M row-major, ldm)
__device__ __forceinline__ v2f frag_at(const float* M, int ldm, int m0, int k0) {
  int lane = threadIdx.x & 31;
  int half = lane >> 4, m = lane & 15;
  const float* p = M + (k0 + half * 2) * ldm + m0 + m;
  v2f r; r.x = p[0]; r.y = p[ldm]; return r;
}

// B fragment: B[k][n], row-major source (ldb). N = lane&15, K split by half.
__device__ __forceinline__ v2f frag_b(const float* B, int ldb, int k0, int n0) {
  int lane = threadIdx.x & 31;
  int half = lane >> 4, n = lane & 15;
  const float* p = B + (k0 + half * 2) * ldb + n0 + n;
  v2f r; r.x = p[0]; r.y = p[ldb]; return r;
}

// B fragment from transposed source: B[k][n] = M[n][k] (M row-major, ldm)
__device__ __forceinline__ v2f frag_bt(const float* M, int ldm, int k0, int n0) {
  int lane = threadIdx.x & 31;
  int half = lane >> 4, n = lane & 15;
  const float* p = M + (n0 + n) * ldm + k0 + half * 2;
  v2f r; r.x = p[0]; r.y = p[1]; return r;
}

// Synthetic lower-triangular-ones A fragment: A[m][k] = (m0+m >= k0+k)
__device__ __forceinline__ v2f frag_tri(int m0, int k0) {
  int lane = threadIdx.x & 31;
  int half = lane >> 4, m = lane & 15;
  int gm = m0 + m, gk = k0 + half * 2;
  v2f r;
  r.x = (gm >= gk) ? 1.0f : 0.0f;
  r.y = (gm >= gk + 1) ? 1.0f : 0.0f;
  return r;
}

// C/D tile store: row = m0 + half*8 + r, col = n0 + (lane&15)
__device__ __forceinline__ void store_c(float* C, int ldc, int m0, int n0, v8f c) {
  int lane = threadIdx.x & 31;
  int half = lane >> 4, n = lane & 15;
  float* p = C + (m0 + half * 8) * ldc + n0 + n;
#pragma unroll
  for (int r = 0; r < 8; ++r) p[r * ldc] = c[r];
}

// -------- CDNA5 async global->LDS copy (ASYNCcnt path), b128 per lane ------
__device__ __forceinline__ void async_g2l_b128(void* lds, const void* g) {
#if __has_builtin(__builtin_amdgcn_global_load_async_to_lds_b128)
  typedef __attribute__((address_space(1))) v4i* gp_t;
  typedef __attribute__((address_space(3))) v4i* lp_t;
  // AS1 pointer: same 64-bit value. AS3 pointer: low 32 bits of the flat
  // shared address are the LDS byte offset (aperture rule); AS3 is 32-bit.
  __builtin_amdgcn_global_load_async_to_lds_b128(
      (gp_t)(uintptr_t)g, (lp_t)(uint32_t)(uintptr_t)lds, 0, 0);
#else
  uint32_t l = (uint32_t)(uintptr_t)lds;
  asm volatile("global_load_async_to_lds_b128 %0, %1, off"
               :: "v"(l), "v"(g)
               : "memory");
#endif
}

__device__ __forceinline__ void wait_async() {
#if __has_builtin(__builtin_amdgcn_s_wait_asynccnt)
  __builtin_amdgcn_s_wait_asynccnt(0);
#else
  asm volatile("s_wait_asynccnt 0" ::: "memory");
#endif
}

// ---------------- Problem constants ----------------
#define BSZ 2
#define HEADS 8
#define SEQ 1024
#define DM 512
#define DEP 64      // head depth == n_features
#define BH (BSZ * HEADS)
#define CHUNK 64
#define NCH (SEQ / CHUNK)   // 16
#define LDP 68              // padded LDS stride: 272B rows -> 16B aligned

// ---------------- Kernel 0: fold ORF into projection weights ----------------
// Mq[f][d] = norm * sum_e orf_q[f][e] * Wq[e][d] ; cq[f] = norm * orf_q[f]·bq
__global__ void k_prep(const float* __restrict__ orfq, const float* __restrict__ Wq,
                       const float* __restrict__ bq,
                       const float* __restrict__ orfk, const float* __restrict__ Wk,
                       const float* __restrict__ bk,
                       float* __restrict__ Mq, float* __restrict__ cq,
                       float* __restrict__ Mk, float* __restrict__ ck) {
  const float norm = 0.35355339059327373f;  // 1/64^0.25
  int tid = threadIdx.x;
  for (int i = tid; i < DEP * DEP; i += 256) {
    int f = i >> 6, d = i & 63;
    float sq = 0.f, sk = 0.f;
    for (int e = 0; e < DEP; ++e) {
      sq += orfq[f * DEP + e] * Wq[e * DEP + d];
      sk += orfk[f * DEP + e] * Wk[e * DEP + d];
    }
    Mq[i] = norm * sq;
    Mk[i] = norm * sk;
  }
  if (tid < DEP) {
    float sq = 0.f, sk = 0.f;
    for (int e = 0; e < DEP; ++e) {
      sq += orfq[tid * DEP + e] * bq[e];
      sk += orfk[tid * DEP + e] * bk[e];
    }
    cq[tid] = norm * sq;
    ck[tid] = norm * sk;
  }
}

// ---------------- Kernel 1: per-head GEMM + optional relu feature map ------
// Y[bh][s][n] = act( sum_d X[b][s][h*64+d] * M[n][d] + c[n] )
__global__ void __launch_bounds__(128)
k_proj(const float* __restrict__ X, const float* __restrict__ M,
       const float* __restrict__ bias, float* __restrict__ Y, int act) {
  __shared__ __align__(16) float Xt[CHUNK][LDP];
  int bh = blockIdx.x;
  int b = bh >> 3, h = bh & 7;
  int s0 = blockIdx.y * CHUNK;
  int tid = threadIdx.x;

  for (int i = tid * 4; i < CHUNK * DEP; i += 128 * 4) {
    int r = i >> 6, c = i & 63;
    async_g2l_b128(&Xt[r][c],
                   X + ((size_t)(b * SEQ + s0 + r)) * DM + h * DEP + c);
  }
  wait_async();
  __syncthreads();

  int wid = tid >> 5;
  int lane = tid & 31;
  int m0 = wid * 16;
  float* Yb = Y + ((size_t)bh * SEQ + s0) * DEP;

#pragma unroll
  for (int j = 0; j < 4; ++j) {
    int n0 = j * 16;
    v8f acc = {};
#pragma unroll
    for (int k0 = 0; k0 < DEP; k0 += 4) {
      v2f a = frag_a(&Xt[0][0], LDP, m0, k0);
      v2f bb = frag_bt(M, DEP, k0, n0);  // B[k=d][n=f] = M[f][d]
      acc = wmma4(a, bb, acc);
    }
    float bi = bias[n0 + (lane & 15)];
    int half = lane >> 4, n = lane & 15;
    float* p = Yb + (m0 + half * 8) * DEP + n0 + n;
#pragma unroll
    for (int r = 0; r < 8; ++r) {
      float v = acc[r] + bi;
      if (act) v = fmaxf(v, 0.0f) + 0.001f;
      p[r * DEP] = v;
    }
  }
}

// ---------------- Kernel 2: per-chunk partial sums KV_c = K^T V, Ks_c ------
__global__ void __launch_bounds__(128)
k_chunksum(const float* __restrict__ phik, const float* __restrict__ vproj,
           float* __restrict__ KV, float* __restrict__ Ks) {
  __shared__ __align__(16) float Kc[CHUNK][LDP];
  __shared__ __align__(16) float Vc[CHUNK][LDP];
  int bh = blockIdx.x, ch = blockIdx.y;
  int s0 = ch * CHUNK;
  int tid = threadIdx.x;
  const float* Kb = phik + ((size_t)bh * SEQ + s0) * DEP;
  const float* Vb = vproj + ((size_t)bh * SEQ + s0) * DEP;

  for (int i = tid * 4; i < CHUNK * DEP; i += 128 * 4) {
    int r = i >> 6, c = i & 63;
    async_g2l_b128(&Kc[r][c], Kb + r * DEP + c);
    async_g2l_b128(&Vc[r][c], Vb + r * DEP + c);
  }
  wait_async();
  __syncthreads();

  int wid = tid >> 5;
  int m0 = wid * 16;  // f-tile
  float* KVo = KV + ((size_t)bh * NCH + ch) * DEP * DEP;
#pragma unroll
  for (int j = 0; j < 4; ++j) {
    int n0 = j * 16;  // d-tile
    v8f acc = {};
#pragma unroll
    for (int k0 = 0; k0 < CHUNK; k0 += 4) {
      v2f a = frag_at(&Kc[0][0], LDP, m0, k0);  // A[f][s] = Kc[s][f]
      v2f bb = frag_b(&Vc[0][0], LDP, k0, n0);  // B[s][d]
      acc = wmma4(a, bb, acc);
    }
    store_c(KVo, DEP, m0, n0, acc);
  }
  if (tid < DEP) {
    float s = 0.f;
    for (int r = 0; r < CHUNK; ++r) s += Kc[r][tid];
    Ks[((size_t)bh * NCH + ch) * DEP + tid] = s;
  }
}

// ---------------- Kernel 3: exclusive prefix scan over chunks --------------
__global__ void __launch_bounds__(256)
k_scan(float* __restrict__ KV, float* __restrict__ Ks) {
  int bh = blockIdx.x;
  int tid = threadIdx.x;
  for (int i = tid; i < DEP * DEP; i += 256) {
    float run = 0.f;
    for (int c = 0; c < NCH; ++c) {
      float* p = KV + ((size_t)bh * NCH + c) * DEP * DEP + i;
      float t = *p;
      *p = run;
      run += t;
    }
  }
  if (tid < DEP) {
    float run = 0.f;
    for (int c = 0; c < NCH; ++c) {
      float* p = Ks + ((size_t)bh * NCH + c) * DEP + tid;
      float t = *p;
      *p = run;
      run += t;
    }
  }
}

// ---------------- Kernel 4: intra-chunk causal attention -------------------
// Z = prefZ + L@K ; qn = q / Z ; O = qn@P + tril(qn@K^T)@V
__global__ void __launch_bounds__(128)
k_attn(const float* __restrict__ phiq, const float* __restrict__ phik,
       const float* __restrict__ vproj, const float* __restrict__ KV,
       const float* __restrict__ Ks, float* __restrict__ O) {
  __shared__ __align__(16) float Kc[CHUNK][LDP];
  __shared__ __align__(16) float Qn[CHUNK][LDP];
  __shared__ __align__(16) float At[CHUNK][LDP];
  int bh = blockIdx.x, ch = blockIdx.y;
  int s0 = ch * CHUNK;
  int tid = threadIdx.x;
  int wid = tid >> 5, lane = tid & 31;
  int half = lane >> 4, nn = lane & 15;
  int m0 = wid * 16;

  const float* Kb = phik + ((size_t)bh * SEQ + s0) * DEP;
  const float* Qb = phiq + ((size_t)bh * SEQ + s0) * DEP;
  const float* Vb = vproj + ((size_t)bh * SEQ + s0) * DEP;
  const float* P = KV + ((size_t)bh * NCH + ch) * DEP * DEP;   // exclusive
  const float* pz = Ks + ((size_t)bh * NCH + ch) * DEP;        // exclusive

  for (int i = tid * 4; i < CHUNK * DEP; i += 128 * 4) {
    int r = i >> 6, c = i & 63;
    async_g2l_b128(&Kc[r][c], Kb + r * DEP + c);
  }
  wait_async();
  __syncthreads();

  // Phase 1: Z = prefZ + L@K (L synthesized in registers), then qn = q/Z
#pragma unroll
  for (int j = 0; j < 4; ++j) {
    int n0 = j * 16;  // f-tile
    v8f acc = {};
#pragma unroll
    for (int k0 = 0; k0 < CHUNK; k0 += 4) {
      v2f a = frag_tri(m0, k0);
      v2f bb = frag_b(&Kc[0][0], LDP, k0, n0);
      acc = wmma4(a, bb, acc);
    }
    float zpre = pz[n0 + nn];
    int f = n0 + nn;
#pragma unroll
    for (int r = 0; r < 8; ++r) {
      int srow = m0 + half * 8 + r;
      float z = acc[r] + zpre;
      float q = Qb[srow * DEP + f];
      Qn[srow][f] = q / z;
    }
  }
  __syncthreads();

  // Phase 2: out1 = qn @ P   (P[f][d] from global, ld=64)
  v8f accO[4];
#pragma unroll
  for (int j = 0; j < 4; ++j) {
    int n0 = j * 16;  // d-tile
    v8f acc = {};
#pragma unroll
    for (int k0 = 0; k0 < DEP; k0 += 4) {
      v2f a = frag_a(&Qn[0][0], LDP, m0, k0);
      v2f bb = frag_b(P, DEP, k0, n0);
      acc = wmma4(a, bb, acc);
    }
    accO[j] = acc;
  }

  // Phase 3: Attn = tril(qn @ K^T) -> LDS (each wave owns its 16 rows)
#pragma unroll
  for (int j = 0; j < 4; ++j) {
    int n0 = j * 16;  // t-tile
    v8f acc = {};
#pragma unroll
    for (int k0 = 0; k0 < DEP; k0 += 4) {
      v2f a = frag_a(&Qn[0][0], LDP, m0, k0);
      v2f bb = frag_bt(&Kc[0][0], LDP, k0, n0);  // B[f][t] = Kc[t][f]
      acc = wmma4(a, bb, acc);
    }
    int t = n0 + nn;
#pragma unroll
    for (int r = 0; r < 8; ++r) {
      int srow = m0 + half * 8 + r;
      At[srow][t] = (t <= srow) ? acc[r] : 0.0f;
    }
  }
  __syncthreads();

  // Phase 4: out += Attn @ V  (V from global), store O
  float* Ob = O + ((size_t)bh * SEQ + s0) * DEP;
#pragma unroll
  for (int j = 0; j < 4; ++j) {
    int n0 = j * 16;  // d-tile
    v8f acc = accO[j];
#pragma unroll
    for (int k0 = 0; k0 < CHUNK; k0 += 4) {
      v2f a = frag_a(&At[0][0], LDP, m0, k0);
      v2f bb = frag_b(Vb, DEP, k0, n0);
      acc = wmma4(a, bb, acc);
    }
    store_c(Ob, DEP, m0, n0, acc);
  }
}

// ---------------- Kernel 5: final FC, [2048,512] @ Wfc^T + bfc -------------
__global__ void __launch_bounds__(128)
k_fc(const float* __restrict__ O, const float* __restrict__ Wfc,
     const float* __restrict__ bfc, float* __restrict__ out) {
  __shared__ __align__(16) float Atile[CHUNK][LDP];
  __shared__ __align__(16) float BtileT[CHUNK][LDP];  // BtileT[n][k]
  int row0 = blockIdx.x * 64;  // over B*S = 2048
  int col0 = blockIdx.y * 64;  // over DM = 512
  int tid = threadIdx.x;
  int wid = tid >> 5, lane = tid & 31;
  int half = lane >> 4, nn = lane & 15;
  int m0 = wid * 16;

  v8f acc[4] = {};
  for (int kb = 0; kb < DM / 64; ++kb) {
    for (int i = tid * 4; i < 64 * 64; i += 128 * 4) {
      int r = i >> 6, c = i & 63;
      int row = row0 + r;
      int b = row >> 10, s = row & 1023;
      int e = kb * 64 + c;
      int h = e >> 6, d = e & 63;
      async_g2l_b128(&Atile[r][c],
                     O + (((size_t)(b * HEADS + h)) * SEQ + s) * DEP + d);
      // BtileT[n=r][k=c] = Wfc[col0+r][kb*64+c]  (k contiguous in global)
      async_g2l_b128(&BtileT[r][c],
                     Wfc + (size_t)(col0 + r) * DM + kb * 64 + c);
    }
    wait_async();
    __syncthreads();
#pragma unroll
    for (int j = 0; j < 4; ++j) {
      int n0 = j * 16;
#pragma unroll
      for (int k0 = 0; k0 < 64; k0 += 4) {
        v2f a = frag_a(&Atile[0][0], LDP, m0, k0);
        v2f bb = frag_bt(&BtileT[0][0], LDP, k0, n0);  // B[k][n] = BtileT[n][k]
        acc[j] = wmma4(a, bb, acc[j]);
      }
    }
    __syncthreads();
  }
#pragma unroll
  for (int j = 0; j < 4; ++j) {
    int n0 = j * 16;
    float bi = bfc[col0 + n0 + nn];
    float* p = out + (size_t)(row0 + m0 + half * 8) * DM + col0 + n0 + nn;
#pragma unroll
    for (int r = 0; r < 8; ++r) p[r * DM] = acc[j][r] + bi;
  }
}

// ---------------- Launcher -------------------------------------------------
extern "C" void kernel_launch(void* const* d_in, const int* in_sizes, int n_in,
                              void* d_out, int out_size, void* d_ws, size_t ws_size,
                              hipStream_t stream) {
  const float* query = (const float*)d_in[0];
  const float* key   = (const float*)d_in[1];
  const float* value = (const float*)d_in[2];
  const float* Wq    = (const float*)d_in[3];
  const float* bq    = (const float*)d_in[4];
  const float* Wk    = (const float*)d_in[5];
  const float* bk    = (const float*)d_in[6];
  const float* Wv    = (const float*)d_in[7];
  const float* bv    = (const float*)d_in[8];
  const float* orfq  = (const float*)d_in[9];
  const float* orfk  = (const float*)d_in[10];
  const float* Wfc   = (const float*)d_in[11];
  const float* bfc   = (const float*)d_in[12];
  float* out = (float*)d_out;

  float* ws = (float*)d_ws;
  float* Mq    = ws;                 // 4096
  float* cq    = Mq + DEP * DEP;     // 64
  float* Mk    = cq + DEP;           // 4096
  float* ck    = Mk + DEP * DEP;     // 64
  float* phiq  = ck + DEP;           // 16*1024*64
  float* phik  = phiq + (size_t)BH * SEQ * DEP;
  float* vproj = phik + (size_t)BH * SEQ * DEP;
  float* KV    = vproj + (size_t)BH * SEQ * DEP;    // 16*16*64*64
  float* Ks    = KV + (size_t)BH * NCH * DEP * DEP; // 16*16*64
  float* Obuf  = Ks + (size_t)BH * NCH * DEP;

  k_prep<<<1, 256, 0, stream>>>(orfq, Wq, bq, orfk, Wk, bk, Mq, cq, Mk, ck);

  dim3 gProj(BH, SEQ / CHUNK);
  k_proj<<<gProj, 128, 0, stream>>>(query, Mq, cq, phiq, 1);
  k_proj<<<gProj, 128, 0, stream>>>(key,   Mk, ck, phik, 1);
  k_proj<<<gProj, 128, 0, stream>>>(value, Wv, bv, vproj, 0);

  dim3 gChunk(BH, NCH);
  k_chunksum<<<gChunk, 128, 0, stream>>>(phik, vproj, KV, Ks);
  k_scan<<<BH, 256, 0, stream>>>(KV, Ks);
  k_attn<<<gChunk, 128, 0, stream>>>(phiq, phik, vproj, KV, Ks, Obuf);

  dim3 gFc((BSZ * SEQ) / 64, DM / 64);
  k_fc<<<gFc, 128, 0, stream>>>(Obuf, Wfc, bfc, out);
}